// GrayMix_35493609734857
// MI455X (gfx1250) — compile-verified
//
#include <hip/hip_runtime.h>

// GrayMix: out[c] = per-patch-mask ? luma(img) : img[c]
// 3x4096x4096 fp32, 16x16 patches, 256x256 patch grid.
// Pure streaming kernel: 384 MiB traffic -> ~17us at 23.3 TB/s HBM.

#define PATCH   16
#define HDIM    4096
#define WDIM    4096
#define NPY     (WDIM / PATCH)      // 256 patch columns
#define ROWV    (WDIM / 4)          // 1024 float4 per row
#define THREADS 256
#define PER_T   (ROWV / THREADS)    // 4 float4 positions per thread per channel

typedef float f4 __attribute__((ext_vector_type(4)));
typedef int   i4 __attribute__((ext_vector_type(4)));

// Address-space-qualified int4 pointers matching the async builtin's signature
// (diagnostic showed param 0 is `int vector_size(16) __device__ *`, i.e. AS1).
typedef __attribute__((address_space(1))) i4* gbl_i4p;
typedef __attribute__((address_space(3))) i4* lds_i4p;

#if defined(__has_builtin)
#if __has_builtin(__builtin_amdgcn_global_load_async_to_lds_b128) && \
    __has_builtin(__builtin_amdgcn_s_wait_asynccnt)
#define GM_HAVE_ASYNC_LDS 1
#endif
#endif
#ifndef GM_HAVE_ASYNC_LDS
#define GM_HAVE_ASYNC_LDS 0
#endif

__global__ __launch_bounds__(THREADS) void GrayMix_kernel(
    const float* __restrict__ img,
    const unsigned char* __restrict__ mask,  // jnp bool -> 1 byte per patch
    float* __restrict__ out)
{
    __shared__ unsigned char lmask[NPY];     // 256 bytes: mask row for this patch-row

    const int row  = blockIdx.x;             // image row 0..4095
    const int prow = row >> 4;               // patch row 0..255
    const int t    = threadIdx.x;

    // ---- Stage this patch-row's 256 mask bytes into LDS (CDNA5 async path) ----
#if GM_HAVE_ASYNC_LDS
    if (t < (NPY / 16)) {                    // 16 lanes x b128 = 256 bytes
        gbl_i4p gsrc = (gbl_i4p)(mask + (size_t)prow * NPY + t * 16);
        lds_i4p ldst = (lds_i4p)(lmask + t * 16);
        __builtin_amdgcn_global_load_async_to_lds_b128(gsrc, ldst, 0, 0);
    }
    __builtin_amdgcn_s_wait_asynccnt(0);
    __syncthreads();
#else
    if (t < (NPY / 4)) {                     // 64 lanes x b32
        ((unsigned int*)lmask)[t] =
            ((const unsigned int*)(mask + (size_t)prow * NPY))[t];
    }
    __syncthreads();
#endif

    // ---- Stream one image row across 3 channels, float4 per step ----
    const size_t plane  = (size_t)HDIM * WDIM;
    const size_t rowOff = (size_t)row * WDIM;

    const f4* __restrict__ rch = (const f4*)(img + rowOff);
    const f4* __restrict__ gch = (const f4*)(img + plane + rowOff);
    const f4* __restrict__ bch = (const f4*)(img + 2 * plane + rowOff);
    f4* __restrict__ o0 = (f4*)(out + rowOff);
    f4* __restrict__ o1 = (f4*)(out + plane + rowOff);
    f4* __restrict__ o2 = (f4*)(out + 2 * plane + rowOff);

#pragma unroll
    for (int k = 0; k < PER_T; ++k) {
        const int p = t + k * THREADS;       // float4 position in row, 0..1023

        // Non-temporal B128 loads: each byte is read exactly once; working set
        // (384 MiB) is 2x the 192 MB L2, so NT avoids thrashing it.
        f4 r = __builtin_nontemporal_load(&rch[p]);
        f4 g = __builtin_nontemporal_load(&gch[p]);
        f4 b = __builtin_nontemporal_load(&bch[p]);

        // 4 consecutive columns share one 16-wide patch -> one mask byte.
        const bool m = lmask[p >> 2] != 0;

        f4 gray = 0.299f * r + 0.587f * g + 0.114f * b;

        f4 vr = m ? gray : r;
        f4 vg = m ? gray : g;
        f4 vb = m ? gray : b;

        __builtin_nontemporal_store(vr, &o0[p]);
        __builtin_nontemporal_store(vg, &o1[p]);
        __builtin_nontemporal_store(vb, &o2[p]);
    }
}

extern "C" void kernel_launch(void* const* d_in, const int* in_sizes, int n_in,
                              void* d_out, int out_size, void* d_ws, size_t ws_size,
                              hipStream_t stream) {
    const float*         img  = (const float*)d_in[0];          // 3*4096*4096 fp32
    const unsigned char* mask = (const unsigned char*)d_in[1];  // 65536 bool bytes
    float*               out  = (float*)d_out;                  // 3*4096*4096 fp32

    (void)in_sizes; (void)n_in; (void)out_size; (void)d_ws; (void)ws_size;

    GrayMix_kernel<<<HDIM, THREADS, 0, stream>>>(img, mask, out);
}